// CorrelationDimensionLoss_3977139716656
// MI455X (gfx1250) — compile-verified
//
#include <hip/hip_runtime.h>

typedef __attribute__((ext_vector_type(2))) float v2f;
typedef __attribute__((ext_vector_type(4))) float v4f;
typedef __attribute__((ext_vector_type(8))) float v8f;

#define K_DIM   32
#define NR      16
// 10 * log2(e): sigmoid sharpness folded into exp2 domain
#define KLOG2E  14.4269504088896340736f

// Workspace layout (d_ws):
//   [0   .. 127]  double gsum[16]        (per-threshold sigmoid sums)
//   [128 .. 135]  unsigned long long cnt (valid pair count)
//   [256 .. ]     float sq[N]            (per-row squared norms)

__global__ void corr_prep_kernel(const float* __restrict__ pts,
                                 float* __restrict__ sq,
                                 double* __restrict__ gsum,
                                 unsigned long long* __restrict__ gcnt,
                                 int N)
{
    int i = blockIdx.x * blockDim.x + threadIdx.x;
    if (blockIdx.x == 0) {
        if (threadIdx.x < NR) gsum[threadIdx.x] = 0.0;
        if (threadIdx.x == NR) *gcnt = 0ull;
    }
    if (i < N) {
        const float* row = pts + (size_t)i * K_DIM;
        float s = 0.0f;
#pragma unroll
        for (int k = 0; k < K_DIM; k += 4) {
            v4f v = *(const v4f*)(row + k);
            s += v.x * v.x + v.y * v.y + v.z * v.z + v.w * v.w;
        }
        sq[i] = s;
    }
}

// Row start offset in packed upper-triangle tile enumeration:
// row t covers tiles (t,t)..(t,nt-1), so S(t) = t*nt - t*(t-1)/2
__device__ __forceinline__ long tri_start(int t, int nt)
{
    return (long)t * nt - ((long)t * (t - 1)) / 2;
}

// One wave32 per 16x16 upper-triangle tile. Gram dot-products via
// v_wmma_f32_16x16x4_f32 (exact f32), then distance + 16 sigmoids per pair.
__global__ void __launch_bounds__(256, 2)
corr_main_kernel(const float* __restrict__ pts,
                 const float* __restrict__ rv_g,
                 const float* __restrict__ sq,
                 double* __restrict__ gsum,
                 unsigned long long* __restrict__ gcnt,
                 int nt)
{
    __shared__ float    lsum[NR];
    __shared__ unsigned lcnt;

    const int tid  = threadIdx.x;
    const int wave = tid >> 5;
    const int lane = tid & 31;

    if (tid < NR) lsum[tid] = 0.0f;
    if (tid == NR) lcnt = 0u;
    __syncthreads();

    // thresholds scaled into exp2 domain: exp(-10(r-d)) = exp2(KLOG2E*d - rvs[r])
    float rvs[NR];
#pragma unroll
    for (int r = 0; r < NR; ++r) rvs[r] = rv_g[r] * KLOG2E;

    const long T   = tri_start(nt, nt);            // nt*(nt+1)/2 tiles
    const long lin = (long)blockIdx.x * 8 + wave;  // packed tile id (wave-uniform)

    float    acc[NR];
#pragma unroll
    for (int r = 0; r < NR; ++r) acc[r] = 0.0f;
    unsigned cnt = 0;

    if (lin < T) {
        // invert lin -> (ti, tj) with ti <= tj via quadratic approx + fixup
        double dn = (double)nt + 0.5;
        int ti = (int)(dn - sqrt(dn * dn - 2.0 * (double)lin));
        if (ti < 0) ti = 0;
        if (ti > nt - 1) ti = nt - 1;
        while (ti > 0 && tri_start(ti, nt) > lin) --ti;
        while (ti < nt - 1 && tri_start(ti + 1, nt) <= lin) ++ti;
        const int tj = ti + (int)(lin - tri_start(ti, nt));

        const int m  = lane & 15;            // row-in-tile (A) / col-in-tile (B)
        const int kh = lane >> 4;            // K-half select per f32 WMMA layout

        // A operand: lane m holds row ti*16+m, K slots {4ks+2kh, 4ks+2kh+1}
        // B operand: lane m holds col tj*16+m, same K slots (B = A^T tile)
        const float* rowA = pts + (size_t)(ti * 16 + m) * K_DIM + 2 * kh;
        const float* rowB = pts + (size_t)(tj * 16 + m) * K_DIM + 2 * kh;

        v8f c = {0.f, 0.f, 0.f, 0.f, 0.f, 0.f, 0.f, 0.f};
#pragma unroll
        for (int ks = 0; ks < 8; ++ks) {
            v2f a = *(const v2f*)(rowA + 4 * ks);
            v2f b = *(const v2f*)(rowB + 4 * ks);
            c = __builtin_amdgcn_wmma_f32_16x16x4_f32(
                    false, a, false, b, (short)0, c, false, false);
        }

        // squared norms: this lane covers rows ti*16 + 8*kh + [0..7], col tj*16+m
        v4f sqa0 = *(const v4f*)(sq + ti * 16 + 8 * kh);
        v4f sqa1 = *(const v4f*)(sq + ti * 16 + 8 * kh + 4);
        float sqa[8] = {sqa0.x, sqa0.y, sqa0.z, sqa0.w,
                        sqa1.x, sqa1.y, sqa1.z, sqa1.w};
        const float sqb = sq[tj * 16 + m];
        const int   gj  = tj * 16 + m;

#pragma unroll
        for (int v = 0; v < 8; ++v) {
            const int gi = ti * 16 + 8 * kh + v;  // C layout: M = v + 8*kh
            if (gi < gj) {                        // always true off-diagonal
                float d2 = sqa[v] + sqb - 2.0f * c[v];
                d2 = fmaxf(d2, 0.0f);
                float d = __builtin_amdgcn_sqrtf(d2);
                if (d > 0.0f) {
                    cnt++;
                    const float a10 = d * KLOG2E;
#pragma unroll
                    for (int r = 0; r < NR; ++r) {
                        // sigmoid(10(r-d)) = 1/(1+exp2(a10 - rvs[r]))
                        float e = __builtin_amdgcn_exp2f(a10 - rvs[r]);
                        acc[r] += __builtin_amdgcn_rcpf(1.0f + e);
                    }
                }
            }
        }
    }

    // wave32 shuffle reduction, then LDS atomics, then device atomics (f64)
#pragma unroll
    for (int r = 0; r < NR; ++r) {
        float v = acc[r];
        v += __shfl_down(v, 16, 32);
        v += __shfl_down(v, 8, 32);
        v += __shfl_down(v, 4, 32);
        v += __shfl_down(v, 2, 32);
        v += __shfl_down(v, 1, 32);
        if (lane == 0 && v != 0.0f) atomicAdd(&lsum[r], v);
    }
    {
        unsigned v = cnt;
        v += __shfl_down(v, 16, 32);
        v += __shfl_down(v, 8, 32);
        v += __shfl_down(v, 4, 32);
        v += __shfl_down(v, 2, 32);
        v += __shfl_down(v, 1, 32);
        if (lane == 0 && v) atomicAdd(&lcnt, v);
    }
    __syncthreads();

    if (tid < NR) {
        float s = lsum[tid];
        if (s != 0.0f) atomicAdd(&gsum[tid], (double)s);
    }
    if (tid == NR) {
        unsigned c = lcnt;
        if (c) atomicAdd(gcnt, (unsigned long long)c);
    }
}

// Final log-log least-squares fit: out = -slope
__global__ void corr_fit_kernel(const float* __restrict__ rv,
                                const double* __restrict__ gsum,
                                const unsigned long long* __restrict__ gcnt,
                                float* __restrict__ out)
{
    if (blockIdx.x == 0 && threadIdx.x == 0) {
        const double n  = (double)NR;
        const double cn = (double)(*gcnt);
        double sx = 0.0, sy = 0.0, sxx = 0.0, sxy = 0.0;
        for (int r = 0; r < NR; ++r) {
            double x = log((double)rv[r]);
            double y = log(gsum[r] / cn);
            sx += x; sy += y; sxx += x * x; sxy += x * y;
        }
        double slope = (n * sxy - sx * sy) / (n * sxx - sx * sx);
        out[0] = (float)(-slope);
    }
}

extern "C" void kernel_launch(void* const* d_in, const int* in_sizes, int n_in,
                              void* d_out, int out_size, void* d_ws, size_t ws_size,
                              hipStream_t stream)
{
    const float* pts = (const float*)d_in[0];   // [N,32] f32
    const float* rv  = (const float*)d_in[1];   // [16]   f32

    const int N  = in_sizes[0] / K_DIM;         // 8192
    const int nt = N / 16;                      // 512 tiles per dim

    char* ws = (char*)d_ws;
    double*             gsum = (double*)ws;
    unsigned long long* gcnt = (unsigned long long*)(ws + 128);
    float*              sq   = (float*)(ws + 256);

    corr_prep_kernel<<<(N + 255) / 256, 256, 0, stream>>>(pts, sq, gsum, gcnt, N);

    const long totalTiles = (long)nt * (nt + 1) / 2;  // upper triangle only
    const int  blocks     = (int)((totalTiles + 7) / 8);
    corr_main_kernel<<<blocks, 256, 0, stream>>>(pts, rv, sq, gsum, gcnt, nt);

    corr_fit_kernel<<<1, 32, 0, stream>>>(rv, gsum, gcnt, (float*)d_out);
}